// GNN_10385230922554
// MI455X (gfx1250) — compile-verified
//
#include <hip/hip_runtime.h>

typedef __attribute__((ext_vector_type(2))) float v2f;
typedef __attribute__((ext_vector_type(8))) float v8f;

#define IN_C  512
#define HID_C 128
#define OUT_C 40

// LDS row strides chosen so that stride % 64 == 4 (dwords):
// A-fragment ds_load_b64 (lane: M=lane&15, Kpair=(lane>>4)*2) then tiles all
// 64 LDS banks conflict-free across the wave.
#define LDS_S1 516   // 512 + 4
#define LDS_S2 132   // 128 + 4

// ---------------- utility kernels ----------------

__global__ void k_zero4(float4* __restrict__ p, long n4) {
  long i = (long)blockIdx.x * blockDim.x + threadIdx.x;
  long stride = (long)gridDim.x * blockDim.x;
  for (; i < n4; i += stride) p[i] = make_float4(0.f, 0.f, 0.f, 0.f);
}

__global__ void k_degree(const int* __restrict__ dst, float* __restrict__ deg, int E) {
  int i = blockIdx.x * blockDim.x + threadIdx.x;
  if (i < E) atomicAdd(&deg[dst[i]], 1.0f);
}

__global__ void k_dinv(float* __restrict__ deg, int N) {
  int i = blockIdx.x * blockDim.x + threadIdx.x;
  if (i < N) deg[i] = rsqrtf(fmaxf(deg[i], 1.0f));
}

__global__ void k_edgew(const int* __restrict__ src, const int* __restrict__ dst,
                        const float* __restrict__ dinv, float* __restrict__ w, int E) {
  int i = blockIdx.x * blockDim.x + threadIdx.x;
  if (i < E) w[i] = dinv[src[i]] * dinv[dst[i]];
}

__global__ void k_bias_relu(float* __restrict__ h, const float* __restrict__ b, int n) {
  int i = blockIdx.x * blockDim.x + threadIdx.x;
  if (i < n) {
    float v = h[i] + b[i & (HID_C - 1)];
    h[i] = v > 0.f ? v : 0.f;
  }
}

__global__ void k_bias_out(float* __restrict__ o, const float* __restrict__ b, int n) {
  int i = blockIdx.x * blockDim.x + threadIdx.x;
  if (i < n) o[i] += b[i % OUT_C];
}

// ---------------- GEMM1: H0[N,128] = X[N,512] @ W0[128,512]^T ----------------
// Block = 256 threads = 8 waves; block handles one 16-row M-tile, wave w handles
// N-tile w (full 128-wide output row-block). X tile staged in LDS once per block.

__global__ __launch_bounds__(256) void k_gemm1_wmma(
    const float* __restrict__ X, const float* __restrict__ W0,
    float* __restrict__ H, int n_rows) {
  __shared__ float xs[16 * LDS_S1];
  const int m0 = blockIdx.x * 16;
  const int tid = threadIdx.x;
  const bool full_tile = (m0 + 16 <= n_rows);   // uniform per block

  // stage X[m0:m0+16, 0:512] -> LDS (coalesced float4 loads)
  if (full_tile) {
    for (int i = tid; i < 16 * (IN_C / 4); i += 256) {
      int row = i >> 7;          // 128 float4 per row
      int c4  = i & 127;
      float4 v = *((const float4*)(X + (size_t)(m0 + row) * IN_C) + c4);
      *(float4*)(xs + row * LDS_S1 + c4 * 4) = v;
    }
  } else {
    for (int i = tid; i < 16 * (IN_C / 4); i += 256) {
      int row = i >> 7;
      int c4  = i & 127;
      float4 v = make_float4(0.f, 0.f, 0.f, 0.f);
      if (m0 + row < n_rows)
        v = *((const float4*)(X + (size_t)(m0 + row) * IN_C) + c4);
      *(float4*)(xs + row * LDS_S1 + c4 * 4) = v;
    }
  }
  __syncthreads();

  const int lane  = tid & 31;
  const int wave  = tid >> 5;          // 0..7 -> N-tile
  const int mfrag = lane & 15;         // A: row M / B: col N
  const int kgrp  = (lane >> 4) << 1;  // K offset {0,2}
  const int n0    = wave * 16;

  const float* arow = xs + mfrag * LDS_S1 + kgrp;
  const float* brow = W0 + (size_t)(n0 + mfrag) * IN_C + kgrp;

  v8f acc = {};
#pragma unroll 8
  for (int k = 0; k < IN_C; k += 4) {
    v2f a = *(const v2f*)(arow + k);                 // ds_load_2addr_b64, bank-conflict-free
    v2f b = *(const v2f*)(brow + k);                 // global_load_b64, L2-hot
    acc = __builtin_amdgcn_wmma_f32_16x16x4_f32(
        false, a, false, b, (short)0, acc, false, false);
  }

  // D layout: VGPR r -> M = r + 8*(lane>>4), N = lane&15
  const int mbase = m0 + ((lane >> 4) << 3);
  const int ncol  = n0 + (lane & 15);
  float* hp = H + (size_t)mbase * HID_C + ncol;
  if (full_tile) {
#pragma unroll
    for (int r = 0; r < 8; ++r) hp[(size_t)r * HID_C] = acc[r];
  } else {
#pragma unroll
    for (int r = 0; r < 8; ++r)
      if (mbase + r < n_rows) hp[(size_t)r * HID_C] = acc[r];
  }
}

// ---------------- GEMM2: H3[N,40] = H[N,128] @ W1[40,128]^T ----------------
// Block = 96 threads = 3 waves = 3 N-tiles (N padded 40->48, stores guarded).

__global__ __launch_bounds__(96) void k_gemm2_wmma(
    const float* __restrict__ H, const float* __restrict__ W1,
    float* __restrict__ Y, int n_rows) {
  __shared__ float xs[16 * LDS_S2];
  const int m0 = blockIdx.x * 16;
  const int tid = threadIdx.x;
  const bool full_tile = (m0 + 16 <= n_rows);

  if (full_tile) {
    for (int i = tid; i < 16 * (HID_C / 4); i += 96) {
      int row = i >> 5;          // 32 float4 per row
      int c4  = i & 31;
      float4 v = *((const float4*)(H + (size_t)(m0 + row) * HID_C) + c4);
      *(float4*)(xs + row * LDS_S2 + c4 * 4) = v;
    }
  } else {
    for (int i = tid; i < 16 * (HID_C / 4); i += 96) {
      int row = i >> 5;
      int c4  = i & 31;
      float4 v = make_float4(0.f, 0.f, 0.f, 0.f);
      if (m0 + row < n_rows)
        v = *((const float4*)(H + (size_t)(m0 + row) * HID_C) + c4);
      *(float4*)(xs + row * LDS_S2 + c4 * 4) = v;
    }
  }
  __syncthreads();

  const int lane  = tid & 31;
  const int wave  = tid >> 5;          // 0..2
  const int mfrag = lane & 15;
  const int kgrp  = (lane >> 4) << 1;
  const int n0    = wave * 16;

  const int   nb    = n0 + mfrag;
  const float bmask = (nb < OUT_C) ? 1.0f : 0.0f;   // zero out padded B rows
  const float* arow = xs + mfrag * LDS_S2 + kgrp;
  const float* brow = W1 + (size_t)(nb < OUT_C ? nb : 0) * HID_C + kgrp;

  v8f acc = {};
#pragma unroll
  for (int k = 0; k < HID_C; k += 4) {
    v2f a = *(const v2f*)(arow + k);
    v2f b = *(const v2f*)(brow + k);
    b[0] *= bmask;
    b[1] *= bmask;
    acc = __builtin_amdgcn_wmma_f32_16x16x4_f32(
        false, a, false, b, (short)0, acc, false, false);
  }

  const int ncol  = n0 + (lane & 15);
  const int mbase = m0 + ((lane >> 4) << 3);
  if (ncol < OUT_C) {
    float* yp = Y + (size_t)mbase * OUT_C + ncol;
    if (full_tile) {
#pragma unroll
      for (int r = 0; r < 8; ++r) yp[(size_t)r * OUT_C] = acc[r];
    } else {
#pragma unroll
      for (int r = 0; r < 8; ++r)
        if (mbase + r < n_rows) yp[(size_t)r * OUT_C] = acc[r];
    }
  }
}

// ---------------- edge aggregation (wave per edge) ----------------
// out[dst,:] += w[e] * H[src,:]. Feature matrices are L2-resident (51/16 MB
// vs 192 MB L2), so random gathers + fp32 atomics run at L2 bandwidth.

__global__ __launch_bounds__(256) void k_agg128(
    const float* __restrict__ H, const int* __restrict__ src,
    const int* __restrict__ dst, const float* __restrict__ w,
    float* __restrict__ out, int E) {
  int e = (blockIdx.x * 256 + threadIdx.x) >> 5;
  if (e >= E) return;
  int lane = threadIdx.x & 31;
  int s = src[e], d = dst[e];
  float we = w[e];
  float4 v = *((const float4*)(H + (size_t)s * HID_C) + lane);  // 32 lanes x 4 = 128 ch
  float* dp = out + (size_t)d * HID_C + lane * 4;
  atomicAdd(dp + 0, v.x * we);
  atomicAdd(dp + 1, v.y * we);
  atomicAdd(dp + 2, v.z * we);
  atomicAdd(dp + 3, v.w * we);
}

__global__ __launch_bounds__(256) void k_agg40(
    const float* __restrict__ H, const int* __restrict__ src,
    const int* __restrict__ dst, const float* __restrict__ w,
    float* __restrict__ out, int E) {
  int e = (blockIdx.x * 256 + threadIdx.x) >> 5;
  if (e >= E) return;
  int lane = threadIdx.x & 31;
  int s = src[e], d = dst[e];
  float we = w[e];
  const float* sp = H + (size_t)s * OUT_C;
  float* dp = out + (size_t)d * OUT_C;
  atomicAdd(dp + lane, sp[lane] * we);              // channels 0..31
  if (lane < OUT_C - 32)
    atomicAdd(dp + 32 + lane, sp[32 + lane] * we);  // channels 32..39
}

// ---------------- launch ----------------

extern "C" void kernel_launch(void* const* d_in, const int* in_sizes, int n_in,
                              void* d_out, int out_size, void* d_ws, size_t ws_size,
                              hipStream_t stream) {
  const float* x  = (const float*)d_in[0];
  const int*   ei = (const int*)d_in[1];   // edge_index, [2,E] flat (int per contract)
  const float* W0 = (const float*)d_in[2];
  const float* b0 = (const float*)d_in[3];
  const float* W1 = (const float*)d_in[4];
  const float* b1 = (const float*)d_in[5];
  float* out = (float*)d_out;

  const int N = in_sizes[0] / IN_C;
  const int E = in_sizes[1] / 2;
  const int* srcI = ei;
  const int* dstI = ei + E;

  // workspace carve-out (256B aligned slabs)
  char* ws = (char*)d_ws;
  size_t off = 0;
  auto carve = [&](size_t bytes) -> float* {
    float* p = (float*)(ws + off);
    off = (off + bytes + 255) & ~(size_t)255;
    return p;
  };
  float* deg = carve((size_t)N * 4);            // degree -> dinv (in place)
  float* wE  = carve((size_t)E * 4);            // per-edge norm weight
  float* h0  = carve((size_t)N * HID_C * 4);    // X @ W0^T
  float* h1  = carve((size_t)N * HID_C * 4);    // aggregated, then relu+b0 in place
  float* h3  = carve((size_t)N * OUT_C * 4);    // h1 @ W1^T
  (void)ws_size; (void)n_in; (void)out_size;

  const int T = 256;
  auto grid1 = [&](long n) { return dim3((unsigned)((n + T - 1) / T)); };

  // zero accumulators (every call: deterministic across graph replays)
  {
    long n4;
    n4 = (N + 3) / 4;
    k_zero4<<<dim3((unsigned)min((n4 + T - 1) / T, (long)16384)), T, 0, stream>>>((float4*)deg, n4);
    n4 = (long)N * HID_C / 4;
    k_zero4<<<dim3((unsigned)min((n4 + T - 1) / T, (long)16384)), T, 0, stream>>>((float4*)h1, n4);
    n4 = (long)N * OUT_C / 4;
    k_zero4<<<dim3((unsigned)min((n4 + T - 1) / T, (long)16384)), T, 0, stream>>>((float4*)out, n4);
  }

  // degrees -> dinv -> edge weights
  k_degree<<<grid1(E), T, 0, stream>>>(dstI, deg, E);
  k_dinv<<<grid1(N), T, 0, stream>>>(deg, N);
  k_edgew<<<grid1(E), T, 0, stream>>>(srcI, dstI, deg, wE, E);

  // layer 0: project (WMMA) -> aggregate -> bias+relu
  k_gemm1_wmma<<<dim3((N + 15) / 16), 256, 0, stream>>>(x, W0, h0, N);
  k_agg128<<<dim3((E + 7) / 8), 256, 0, stream>>>(h0, srcI, dstI, wE, h1, E);
  k_bias_relu<<<grid1((long)N * HID_C), T, 0, stream>>>(h1, b0, N * HID_C);

  // layer 1: project (WMMA) -> aggregate -> bias
  k_gemm2_wmma<<<dim3((N + 15) / 16), 96, 0, stream>>>(h1, W1, h3, N);
  k_agg40<<<dim3((E + 7) / 8), 256, 0, stream>>>(h3, srcI, dstI, wE, out, E);
  k_bias_out<<<grid1((long)N * OUT_C), T, 0, stream>>>(out, b1, N * OUT_C);
}